// myGCN_11441792876722
// MI455X (gfx1250) — compile-verified
//
#include <hip/hip_runtime.h>
#include <math.h>

typedef __attribute__((ext_vector_type(2))) float v2f;
typedef __attribute__((ext_vector_type(8))) float v8f;

#define HDIM 128
#define GGRP 64

__device__ __forceinline__ float gelu_exact(float x) {
  // 0.5*x*(1+erf(x/sqrt(2)))  (approximate=False)
  return 0.5f * x * (1.0f + erff(x * 0.70710678118654752f));
}

// ---------------- degree / norm ----------------
__global__ void k_init_deg(float* deg, int n) {
  int i = blockIdx.x * blockDim.x + threadIdx.x;
  if (i < n) deg[i] = 1.0f;  // self-loop contributes 1
}
__global__ void k_accum_deg(const int* __restrict__ dst, float* deg, int e) {
  int i = blockIdx.x * blockDim.x + threadIdx.x;
  if (i < e) unsafeAtomicAdd(&deg[dst[i]], 1.0f);
}
__global__ void k_rsqrt_deg(float* deg, int n) {
  int i = blockIdx.x * blockDim.x + threadIdx.x;
  if (i < n) deg[i] = rsqrtf(deg[i]);  // deg >= 1 always (self-loop)
}

// ---------------- GEMM: Y[M,128] = X[M,128] @ W[128,128] via V_WMMA_F32_16X16X4_F32 ----
// One block = 8 waves = 128 output rows. W staged in LDS pre-swizzled into
// B-fragment order so each lane's (b0,b1) pair is one ds_load_b64:
//   for kblk = k/4, half in {0,1}, col c:  pair p = kblk*256 + half*128 + c
//     sW[2p+0] = W[4*kblk + half    ][c]   (B VGPR0: row K = 2*half? no -> K=half)
//     sW[2p+1] = W[4*kblk + 2 + half][c]   (B VGPR1: row K = 2+half)
__global__ __launch_bounds__(256) void k_gemm128(const float* __restrict__ X,
                                                 const float* __restrict__ W,
                                                 float* __restrict__ Y, int nrows) {
  __shared__ float sW[HDIM * HDIM];
  const int tid  = threadIdx.x;
  const int wave = tid >> 5;   // 0..7
  const int lane = tid & 31;
  const int half = lane >> 4;  // 0 or 1 (half-wave)
  const int r    = lane & 15;
  const int rowBase = blockIdx.x * 128;

  for (int i = tid * 4; i < HDIM * HDIM; i += 256 * 4) {
    float4 w4 = *(const float4*)&W[i];
    int kr   = i >> 7;   // W row (K index)
    int c    = i & 127;  // starting col
    int kblk = kr >> 2;
    int sub  = kr & 3;   // 0->(h0,e0) 1->(h1,e0) 2->(h0,e1) 3->(h1,e1)
    int base = (kblk * 256 + (sub & 1) * 128 + c) * 2 + (sub >> 1);
    sW[base + 0] = w4.x;
    sW[base + 2] = w4.y;
    sW[base + 4] = w4.z;
    sW[base + 6] = w4.w;
  }
  __syncthreads();

  // A-frag: lanes 0-15 and 16-31 both hold rows M=0..15; half-wave selects the
  // K pair (VGPR0=K(2*half), VGPR1=K(2*half+1)). Clamp ragged tail; stores guarded.
  int arow = rowBase + wave * 16 + r;
  if (arow >= nrows) arow = nrows - 1;
  const float* Xrow = X + (size_t)arow * HDIM;

  v8f acc[8] = {};
  for (int kblk = 0; kblk < 32; ++kblk) {
    v2f a = *(const v2f*)&Xrow[kblk * 4 + 2 * half];  // 8B-aligned
    const float* bb = &sW[(kblk * 256 + half * 128) * 2];
#pragma unroll
    for (int n = 0; n < 8; ++n) {
      v2f b = *(const v2f*)&bb[(n * 16 + r) * 2];      // single ds_load_b64
      acc[n] = __builtin_amdgcn_wmma_f32_16x16x4_f32(false, a, false, b,
                                                     (short)0, acc[n], false, false);
    }
  }

#pragma unroll
  for (int n = 0; n < 8; ++n) {
#pragma unroll
    for (int j = 0; j < 8; ++j) {  // D: VGPR j -> M = j + 8*half, N = r
      int gm = rowBase + wave * 16 + j + 8 * half;
      if (gm < nrows) Y[(size_t)gm * HDIM + n * 16 + r] = acc[n][j];
    }
  }
}

// ---------------- self-loop term: out[i,:] = tmp[i,:] * dis[i]^2 ----------------
__global__ void k_selfloop(const float* __restrict__ tmp, const float* __restrict__ dis,
                           float* __restrict__ out, int n) {
  int t = blockIdx.x * blockDim.x + threadIdx.x;  // n*32 threads, float4 each
  int i = t >> 5;
  if (i >= n) return;
  int f = (t & 31) * 4;
  float w = dis[i]; w = w * w;
  float4 v = *(const float4*)&tmp[(size_t)i * HDIM + f];
  float4 o = make_float4(v.x * w, v.y * w, v.z * w, v.w * w);
  *(float4*)&out[(size_t)i * HDIM + f] = o;
}

// ---------------- edge scatter: out[d,:] += tmp[s,:] * dis[s]*dis[d] ----------------
__global__ void k_scatter(const float* __restrict__ tmp, const int* __restrict__ src,
                          const int* __restrict__ dst, const float* __restrict__ dis,
                          float* __restrict__ out, int e) {
  int t = blockIdx.x * blockDim.x + threadIdx.x;  // one wave per edge
  int ei = t >> 5;
  if (ei >= e) return;
  int lane = t & 31;
  int s = src[ei], d = dst[ei];
  float w = dis[s] * dis[d];
  float4 v = *(const float4*)&tmp[(size_t)s * HDIM + lane * 4];
  float* o = &out[(size_t)d * HDIM + lane * 4];
  unsafeAtomicAdd(o + 0, v.x * w);
  unsafeAtomicAdd(o + 1, v.y * w);
  unsafeAtomicAdd(o + 2, v.z * w);
  unsafeAtomicAdd(o + 3, v.w * w);
}

// ---------------- bias + exact GELU (in place) ----------------
__global__ void k_biasgelu(float* __restrict__ h, const float* __restrict__ b, int total) {
  int t = blockIdx.x * blockDim.x + threadIdx.x;
  if (t < total) h[t] = gelu_exact(h[t] + b[t & (HDIM - 1)]);
}

// ---------------- pooling ----------------
__global__ void k_zero(float* p, int n) {
  int i = blockIdx.x * blockDim.x + threadIdx.x;
  if (i < n) p[i] = 0.0f;
}
__global__ void k_poolsum(const float* __restrict__ h, const int* __restrict__ batch,
                          float* __restrict__ pooled, int n) {
  int t = blockIdx.x * blockDim.x + threadIdx.x;  // n*32 threads, float4 each
  int i = t >> 5;
  if (i >= n) return;
  int f = (t & 31) * 4;
  int g = batch[i];
  float4 v = *(const float4*)&h[(size_t)i * HDIM + f];
  float* p = &pooled[(size_t)g * HDIM + f];
  unsafeAtomicAdd(p + 0, v.x);
  unsafeAtomicAdd(p + 1, v.y);
  unsafeAtomicAdd(p + 2, v.z);
  unsafeAtomicAdd(p + 3, v.w);
}
__global__ void k_count(const int* __restrict__ batch, float* __restrict__ cnt, int n) {
  int i = blockIdx.x * blockDim.x + threadIdx.x;
  if (i < n) unsafeAtomicAdd(&cnt[batch[i]], 1.0f);
}

// ---------------- head MLP: mean -> (128x64)+ELU -> (64x1) ----------------
__global__ void k_final(const float* __restrict__ pooled, const float* __restrict__ cnt,
                        const float* __restrict__ lw1, const float* __restrict__ lb1,
                        const float* __restrict__ lw2, const float* __restrict__ lb2,
                        float* __restrict__ out) {
  int g = threadIdx.x;
  if (g >= GGRP) return;
  float c = fmaxf(cnt[g], 1.0f);
  float p[HDIM];
#pragma unroll 4
  for (int k = 0; k < HDIM; ++k) p[k] = pooled[g * HDIM + k] / c;
  float o = lb2[0];
  for (int j = 0; j < GGRP; ++j) {
    float a = lb1[j];
#pragma unroll 4
    for (int k = 0; k < HDIM; ++k) a += p[k] * lw1[k * GGRP + j];
    a = (a > 0.0f) ? a : expm1f(a);  // exact ELU, alpha=1
    o += a * lw2[j];
  }
  out[g] = o;
}

extern "C" void kernel_launch(void* const* d_in, const int* in_sizes, int n_in,
                              void* d_out, int out_size, void* d_ws, size_t ws_size,
                              hipStream_t stream) {
  const float* x   = (const float*)d_in[0];
  const float* W1  = (const float*)d_in[1];
  const float* b1  = (const float*)d_in[2];
  const float* W2  = (const float*)d_in[3];
  const float* b2  = (const float*)d_in[4];
  const float* W3  = (const float*)d_in[5];
  const float* b3  = (const float*)d_in[6];
  const float* lw1 = (const float*)d_in[7];
  const float* lb1 = (const float*)d_in[8];
  const float* lw2 = (const float*)d_in[9];
  const float* lb2 = (const float*)d_in[10];
  const int* eidx  = (const int*)d_in[11];
  const int* batch = (const int*)d_in[12];

  const int N = in_sizes[0] / HDIM;
  const int E = in_sizes[11] / 2;
  const int* src = eidx;
  const int* dst = eidx + E;

  float* ws     = (float*)d_ws;
  float* dis    = ws;                          // N
  float* tmp    = dis + N;                     // N*H
  float* hbuf   = tmp + (size_t)N * HDIM;      // N*H
  float* pooled = hbuf + (size_t)N * HDIM;     // G*H
  float* cnt    = pooled + GGRP * HDIM;        // G

  // degrees -> dis = deg^{-1/2}
  k_init_deg<<<(N + 255) / 256, 256, 0, stream>>>(dis, N);
  k_accum_deg<<<(E + 255) / 256, 256, 0, stream>>>(dst, dis, E);
  k_rsqrt_deg<<<(N + 255) / 256, 256, 0, stream>>>(dis, N);

  const float* Xin[3] = {x, hbuf, hbuf};
  const float* Ws[3]  = {W1, W2, W3};
  const float* bs[3]  = {b1, b2, b3};
  for (int l = 0; l < 3; ++l) {
    k_gemm128<<<(N + 127) / 128, 256, 0, stream>>>(Xin[l], Ws[l], tmp, N);
    k_selfloop<<<((N * 32) + 255) / 256, 256, 0, stream>>>(tmp, dis, hbuf, N);
    k_scatter<<<((E * 32) + 255) / 256, 256, 0, stream>>>(tmp, src, dst, dis, hbuf, E);
    k_biasgelu<<<(N * HDIM + 255) / 256, 256, 0, stream>>>(hbuf, bs[l], N * HDIM);
  }

  // pooling + head
  k_zero<<<(GGRP * HDIM + GGRP + 255) / 256, 256, 0, stream>>>(pooled, GGRP * HDIM + GGRP);
  k_poolsum<<<((N * 32) + 255) / 256, 256, 0, stream>>>(hbuf, batch, pooled, N);
  k_count<<<(N + 255) / 256, 256, 0, stream>>>(batch, cnt, N);
  k_final<<<1, GGRP, 0, stream>>>(pooled, cnt, lw1, lb1, lw2, lb2, (float*)d_out);
}